// TransE_64750926954631
// MI455X (gfx1250) — compile-verified
//
#include <hip/hip_runtime.h>
#include <hip/hip_bf16.h>

typedef __attribute__((ext_vector_type(2))) float v2f;
typedef __attribute__((ext_vector_type(8))) float v8f;

#define EMB   128      // embedding dim
#define TILE  16       // triples per wave-tile (WMMA M/N)
#define ROWP  132      // padded LDS row stride in floats (132 % 64 banks -> stride 4 -> conflict-free b64 reads)
#define WPB   8        // waves per 256-thread block (wave32)
#define MAXSEG 256

__global__ void zero_ws_kernel(float* ws, int n) {
    int i = blockIdx.x * blockDim.x + threadIdx.x;
    if (i < n) ws[i] = 0.0f;
}

__global__ __launch_bounds__(256) void transe_wmma_kernel(
    const float* __restrict__ type_emb,
    const float* __restrict__ rel_emb,
    const int*   __restrict__ dom_ids,
    const int*   __restrict__ ran_ids,
    const int*   __restrict__ seg_ids,
    const int*   __restrict__ rel_labels,
    float* __restrict__ ws_sum,
    float* __restrict__ ws_cnt,
    int T, int nseg)
{
    __shared__ float xt_all[WPB][TILE * ROWP];   // per-wave x = h + r - t tiles
    __shared__ float bin_sum[MAXSEG];
    __shared__ float bin_cnt[MAXSEG];

    const int tid = threadIdx.x;
    for (int i = tid; i < MAXSEG; i += blockDim.x) { bin_sum[i] = 0.0f; bin_cnt[i] = 0.0f; }
    __syncthreads();

    const int wave = tid >> 5;     // wave32
    const int lane = tid & 31;
    float* xt = xt_all[wave];

    const int tiles  = (T + TILE - 1) / TILE;
    const int gwave  = blockIdx.x * WPB + wave;
    const int nwaves = gridDim.x * WPB;

    // A/B operand addressing for V_WMMA_F32_16X16X4_F32:
    // lane m (<16) holds row m, K={0,1}; lane m+16 holds row m, K={2,3}.
    const int arow = lane & 15;
    const int koff = (lane >> 4) << 1;
    const float* abase = xt + arow * ROWP + koff;

    for (int tile = gwave; tile < tiles; tile += nwaves) {
        const int base = tile * TILE;

        // lanes 0..15: load ids for this tile's 16 triples
        const int my   = base + (lane & 15);
        const bool v16 = (lane < TILE) && (my < T);
        int dj = 0, tj = 0, sj = 0, rj = 0;
        if (v16) {
            dj = dom_ids[my];
            tj = ran_ids[my];
            sj = seg_ids[my];
            rj = rel_labels[sj];
        }

        // Build x rows in LDS: one fully-coalesced 512B row gather per id.
        #pragma unroll 4
        for (int j = 0; j < TILE; ++j) {
            const int dd = __shfl(dj, j, 32);
            const int tt = __shfl(tj, j, 32);
            const int rr = __shfl(rj, j, 32);
            float4 x;
            if (base + j < T) {
                const float4 h = *(const float4*)(type_emb + (size_t)dd * EMB + lane * 4);
                const float4 t = *(const float4*)(type_emb + (size_t)tt * EMB + lane * 4);
                const float4 r = *(const float4*)(rel_emb  + (size_t)rr * EMB + lane * 4);
                x = make_float4(h.x + r.x - t.x, h.y + r.y - t.y,
                                h.z + r.z - t.z, h.w + r.w - t.w);
            } else {
                x = make_float4(0.f, 0.f, 0.f, 0.f);
            }
            *(float4*)(xt + j * ROWP + lane * 4) = x;   // conflict-free: consecutive banks
        }
        // Same-wave DS ops are processed in order: reads below see the stores above.

        // D = X * X^T accumulated over K=128 in 32 steps; diagonal = ||x_m||^2.
        v8f c = {0.f, 0.f, 0.f, 0.f, 0.f, 0.f, 0.f, 0.f};
        #pragma unroll
        for (int kc = 0; kc < EMB / 4; ++kc) {
            v2f a = *(const v2f*)(abase + kc * 4);
            c = __builtin_amdgcn_wmma_f32_16x16x4_f32(
                    /*neg_a=*/false, a, /*neg_b=*/false, a,
                    /*c_mod=*/(short)0, c, /*reuse_a=*/false, /*reuse_b=*/false);
        }

        // Extract diagonal: (v,v) lives in VGPR v lane v; (v+8,v+8) in VGPR v lane v+24.
        float s = 0.0f;
        #pragma unroll
        for (int v = 0; v < 8; ++v) {
            const float cv = c[v];
            const float t0 = __shfl(cv, v, 32);
            const float t1 = __shfl(cv, v + 24, 32);
            if (lane == v)     s = t0;
            if (lane == v + 8) s = t1;
        }

        // score = -||x||^2 ; accumulate into LDS segment bins.
        if (v16) {
            atomicAdd(&bin_sum[sj], -s);
            atomicAdd(&bin_cnt[sj], 1.0f);
        }
    }

    __syncthreads();
    // One global f32 atomic per non-empty segment per block.
    for (int i = tid; i < nseg && i < MAXSEG; i += blockDim.x) {
        const float bc = bin_cnt[i];
        if (bc != 0.0f) {
            unsafeAtomicAdd(&ws_sum[i], bin_sum[i]);
            unsafeAtomicAdd(&ws_cnt[i], bc);
        }
    }
}

__global__ void finalize_kernel(const float* __restrict__ ws_sum,
                                const float* __restrict__ ws_cnt,
                                float* __restrict__ out, int nseg) {
    int i = blockIdx.x * blockDim.x + threadIdx.x;
    if (i < nseg) {
        const float c = ws_cnt[i];
        out[i] = (c > 0.0f) ? (ws_sum[i] / c) : 0.0f;
    }
}

extern "C" void kernel_launch(void* const* d_in, const int* in_sizes, int n_in,
                              void* d_out, int out_size, void* d_ws, size_t ws_size,
                              hipStream_t stream) {
    const float* type_emb   = (const float*)d_in[0];
    const float* rel_emb    = (const float*)d_in[1];
    const int*   dom_ids    = (const int*)d_in[2];
    const int*   ran_ids    = (const int*)d_in[3];
    const int*   seg_ids    = (const int*)d_in[4];
    const int*   rel_labels = (const int*)d_in[5];
    float* out = (float*)d_out;

    const int T    = in_sizes[2];   // number of triples
    const int nseg = out_size;      // number of segments (256)

    float* ws_sum = (float*)d_ws;
    float* ws_cnt = ws_sum + nseg;

    const int zn = 2 * nseg;
    zero_ws_kernel<<<(zn + 255) / 256, 256, 0, stream>>>(ws_sum, zn);

    const int blocks = 1024;        // 8192 wave32s; ~7.6 tiles each for T=1M
    transe_wmma_kernel<<<blocks, 256, 0, stream>>>(
        type_emb, rel_emb, dom_ids, ran_ids, seg_ids, rel_labels,
        ws_sum, ws_cnt, T, nseg);

    finalize_kernel<<<(nseg + 255) / 256, 256, 0, stream>>>(ws_sum, ws_cnt, out, nseg);
}